// CameraEstimator_90048284328709
// MI455X (gfx1250) — compile-verified
//
#include <hip/hip_runtime.h>
#include <math.h>

// ---------------------------------------------------------------------------
// CameraEstimator for MI455X (gfx1250, wave32).
//   camera[b] = sum_c x[b,c,:,:] * W[c]        (HBM-bound: 302 MB streamed)
//   out[b]    = closest rotation to camera[b]  (per-lane 3x3 Jacobi SVD)
//
// Contraction on V_WMMA_F32_16X16X4_F32. A-operand = x tile (16 b-rows x 4 K),
// documented f32 A layout (lane=M, VGPR+lane-half=K) -> contiguous float2
// stream per lane. B-operand = sparse weight tile W~[k][n]=(k%9==n)?W[k/9]:0,
// synthesized branch-free: k%9 / k/9 are compile-time per element inside a
// fully-unrolled 36-element (9-WMMA) period; W[4p..4p+3] is wave-uniform and
// comes from one scalar b128 load per period.
// ---------------------------------------------------------------------------

typedef float v2f __attribute__((ext_vector_type(2)));
typedef float v8f __attribute__((ext_vector_type(8)));

#define CCH   256
#define KTOT  (CCH * 9)          // 2304 = contraction length = 64 periods of 36

// ---- constant-index Jacobi rotation (NR convention), fully inlined --------
template <int P, int Q, int R>
__device__ __forceinline__ void jrot(float s[3][3], float v[3][3]) {
  float spq = s[P][Q];
  float spp = s[P][P];
  float sqq = s[Q][Q];
  bool  doit = fabsf(spq) > 1e-30f;
  float tau = (sqq - spp) / (2.0f * spq);
  float tt  = copysignf(1.0f, tau) / (fabsf(tau) + sqrtf(1.0f + tau * tau));
  float t   = doit ? tt : 0.0f;                 // NaN path selected away
  float c   = 1.0f / sqrtf(1.0f + t * t);       // t already safe
  float sn  = t * c;
  s[P][P] = spp - t * spq;
  s[Q][Q] = sqq + t * spq;
  s[P][Q] = 0.0f; s[Q][P] = 0.0f;
  float spr = s[P][R], sqr = s[Q][R];
  float nspr = c * spr - sn * sqr;
  float nsqr = sn * spr + c * sqr;
  s[P][R] = nspr; s[R][P] = nspr;
  s[Q][R] = nsqr; s[R][Q] = nsqr;
#pragma unroll
  for (int i = 0; i < 3; ++i) {
    float vip = v[i][P], viq = v[i][Q];
    v[i][P] = c * vip - sn * viq;
    v[i][Q] = sn * vip + c * viq;
  }
}

__global__ __launch_bounds__(256) void camera_rot_kernel(
    const float* __restrict__ x, const float* __restrict__ W,
    float* __restrict__ out, int Bn) {
  __shared__ float ldsCam[8][16][9];   // [wave][matrix-in-tile][element]

  const int tid   = threadIdx.x;
  const int lane  = tid & 31;
  const int wave  = tid >> 5;
  const int mn    = lane & 15;         // M for A-tile, N for B/C-tile
  const int khalf = lane >> 4;         // K-pair select per ISA f32 A layout
  const bool kh   = (khalf != 0);

  const int b0   = (blockIdx.x * 8 + wave) * 16;
  int brow       = b0 + mn;
  int bclamp     = brow < Bn ? brow : (Bn - 1);
  const float* xb = x + (size_t)bclamp * KTOT + 2 * khalf;

  // ---- K-loop: 64 periods x 9 WMMAs; period body fully unrolled -----------
  v8f acc = {};
  for (int p = 0; p < KTOT / 36; ++p) {
    // 4 wave-uniform weights for this period -> scalar b128 load
    const float4 wq = *(const float4*)(W + 4 * p);
    const float wsel[4] = {wq.x, wq.y, wq.z, wq.w};

#pragma unroll
    for (int i = 0; i < 9; ++i) {
      const int ka = 4 * i;            // K offset in period, khalf=0 element 0
      const int kb = 4 * i + 2;        // K offset in period, khalf=1 element 0
      // element 0 of B operand (VGPR0)
      const int j0  = kh ? (kb % 9) : (ka % 9);            // cndmask of literals
      const float w0 = kh ? wsel[kb / 9] : wsel[ka / 9];   // cndmask of sregs
      // element 1 of B operand (VGPR1)
      const int j1  = kh ? ((kb + 1) % 9) : ((ka + 1) % 9);
      const float w1 = kh ? wsel[(kb + 1) / 9] : wsel[(ka + 1) / 9];

      v2f a = __builtin_nontemporal_load((const v2f*)(xb + 4 * i));
      v2f b;
      b.x = (j0 == mn) ? w0 : 0.0f;
      b.y = (j1 == mn) ? w1 : 0.0f;
      acc = __builtin_amdgcn_wmma_f32_16x16x4_f32(
          /*neg_a=*/false, a, /*neg_b=*/false, b,
          /*c_mod=*/(short)0, acc, /*reuse_a=*/false, /*reuse_b=*/false);
    }
    xb += 36;
  }

  // ---- scatter C tile: VGPR r / lane-half -> M, lane -> N -----------------
  if (mn < 9) {
#pragma unroll
    for (int r = 0; r < 8; ++r) {
      int m = r + 8 * khalf;
      ldsCam[wave][m][mn] = acc[r];
    }
  }
  __syncthreads();

  // ---- per-lane 3x3 closest-rotation (lanes 0-15 of each wave) ------------
  if (lane < 16) {
    int b = b0 + lane;
    if (b < Bn) {
      float A[3][3];
#pragma unroll
      for (int i = 0; i < 3; ++i)
#pragma unroll
        for (int j = 0; j < 3; ++j) A[i][j] = ldsCam[wave][lane][i * 3 + j];

      // S = A^T A
      float s[3][3];
#pragma unroll
      for (int i = 0; i < 3; ++i)
#pragma unroll
        for (int j = 0; j < 3; ++j)
          s[i][j] = A[0][i] * A[0][j] + A[1][i] * A[1][j] + A[2][i] * A[2][j];

      float v[3][3] = {{1.f, 0.f, 0.f}, {0.f, 1.f, 0.f}, {0.f, 0.f, 1.f}};
#pragma unroll
      for (int sweep = 0; sweep < 6; ++sweep) {
        jrot<0, 1, 2>(s, v);
        jrot<0, 2, 1>(s, v);
        jrot<1, 2, 0>(s, v);
      }

      // sort eigenpairs descending, track det(V) parity
      float w0 = s[0][0], w1 = s[1][1], w2 = s[2][2];
      float detV = 1.0f;
      if (w0 < w1) { float t = w0; w0 = w1; w1 = t;
        for (int i = 0; i < 3; ++i) { float tv = v[i][0]; v[i][0] = v[i][1]; v[i][1] = tv; }
        detV = -detV; }
      if (w0 < w2) { float t = w0; w0 = w2; w2 = t;
        for (int i = 0; i < 3; ++i) { float tv = v[i][0]; v[i][0] = v[i][2]; v[i][2] = tv; }
        detV = -detV; }
      if (w1 < w2) { float t = w1; w1 = w2; w2 = t;
        for (int i = 0; i < 3; ++i) { float tv = v[i][1]; v[i][1] = v[i][2]; v[i][2] = tv; }
        detV = -detV; }

      // b_i = A * v_i ; u0,u1 by Gram-Schmidt
      float bc0[3], bc1[3];
#pragma unroll
      for (int i = 0; i < 3; ++i) {
        bc0[i] = A[i][0] * v[0][0] + A[i][1] * v[1][0] + A[i][2] * v[2][0];
        bc1[i] = A[i][0] * v[0][1] + A[i][1] * v[1][1] + A[i][2] * v[2][1];
      }
      float n0 = sqrtf(bc0[0] * bc0[0] + bc0[1] * bc0[1] + bc0[2] * bc0[2]);
      float i0 = 1.0f / fmaxf(n0, 1e-20f);
      float u0[3] = {bc0[0] * i0, bc0[1] * i0, bc0[2] * i0};
      float dp = u0[0] * bc1[0] + u0[1] * bc1[1] + u0[2] * bc1[2];
#pragma unroll
      for (int i = 0; i < 3; ++i) bc1[i] -= dp * u0[i];
      float n1 = sqrtf(bc1[0] * bc1[0] + bc1[1] * bc1[1] + bc1[2] * bc1[2]);
      float i1 = 1.0f / fmaxf(n1, 1e-20f);
      float u1[3] = {bc1[0] * i1, bc1[1] * i1, bc1[2] * i1};
      // third effective column = det(V) * (u0 x u1)
      float cr[3] = {detV * (u0[1] * u1[2] - u0[2] * u1[1]),
                     detV * (u0[2] * u1[0] - u0[0] * u1[2]),
                     detV * (u0[0] * u1[1] - u0[1] * u1[0])};

      // R = u0 v0^T + u1 v1^T + cr v2^T  ==  U diag(1,1,det) V^T
      float* o = out + (size_t)b * 9;
#pragma unroll
      for (int i = 0; i < 3; ++i)
#pragma unroll
        for (int j = 0; j < 3; ++j)
          o[i * 3 + j] = u0[i] * v[j][0] + u1[i] * v[j][1] + cr[i] * v[j][2];
    }
  }
}

extern "C" void kernel_launch(void* const* d_in, const int* in_sizes, int n_in,
                              void* d_out, int out_size, void* d_ws, size_t ws_size,
                              hipStream_t stream) {
  (void)n_in; (void)out_size; (void)d_ws; (void)ws_size;
  const float* x = (const float*)d_in[0];
  const float* W = (const float*)d_in[1];
  float* out = (float*)d_out;
  const int C  = in_sizes[1];                 // 256
  const int Bn = in_sizes[0] / (C * 9);       // 32768
  const int blocks = (Bn + 127) / 128;        // 128 matrices per 256-thread block
  camera_rot_kernel<<<blocks, 256, 0, stream>>>(x, W, out, Bn);
}